// YOLOv1Loss_91233695301881
// MI455X (gfx1250) — compile-verified
//
#include <hip/hip_runtime.h>

// YOLOv1 loss on MI455X (gfx1250): HBM-bound streaming reduction.
// - Coalesced b128 staging of pred records into LDS (records are 120B, so
//   per-lane strided loads would shred coalescing; every 128B line is needed
//   anyway, so staging the full record costs no extra HBM traffic).
// - Targets: only ch4/ch9 needed -> two aligned b64 loads per lane.
// - Wave reduction via V_WMMA_F32_16X16X4_F32 with B = ones (exact fp32).
// - Two-pass deterministic reduction (no float atomics).

typedef float v2f __attribute__((ext_vector_type(2)));
typedef float v4f __attribute__((ext_vector_type(4)));
typedef float v8f __attribute__((ext_vector_type(8)));

#define BLOCK 256
#define NB    512
#define CH    30
#define WAVES (BLOCK / 32)

// jnp.maximum / jnp.max propagate NaN; fmaxf does not.
__device__ __forceinline__ float jmax_nan(float a, float b) {
  if (a != a) return a;
  if (b != b) return b;
  return fmaxf(a, b);
}

// Exact-fp32 cross-lane sum of one value per lane using WMMA f32 16x16x4.
// A[m,0] = lane m partial, A[m,2] = lane m+16 partial (VGPR1 = 0), B = ones:
// D[m,n] = acc[m] + acc[m+16] for every n. Lane 0 holds rows 0..7 of col 0,
// lane 16 holds rows 8..15 of col 0.
__device__ __forceinline__ float wave_sum_wmma(float v) {
  v2f a; a[0] = v;    a[1] = 0.0f;
  v2f b; b[0] = 1.0f; b[1] = 1.0f;
  v8f c = {};
  c = __builtin_amdgcn_wmma_f32_16x16x4_f32(false, a, false, b,
                                            (short)0, c, false, false);
  float s = ((c[0] + c[1]) + (c[2] + c[3])) + ((c[4] + c[5]) + (c[6] + c[7]));
  int si = __builtin_bit_cast(int, s);
  float lo = __builtin_bit_cast(float, __builtin_amdgcn_readlane(si, 0));
  float hi = __builtin_bit_cast(float, __builtin_amdgcn_readlane(si, 16));
  return lo + hi;
}

__global__ void __launch_bounds__(BLOCK) yolo_partial_kernel(
    const float* __restrict__ P, const float* __restrict__ T,
    float* __restrict__ partial, int M)
{
  __shared__ float sp[BLOCK * CH];   // 30 KB staged pred records
  __shared__ float wred[WAVES];
  const int tid = threadIdx.x;
  float acc = 0.0f;

  for (int base = blockIdx.x * BLOCK; base < M; base += NB * BLOCK) {
    __syncthreads();  // previous tile fully consumed
    // ---- coalesced b128 staging of 256 pred records (7680 floats) ----
    {
      const v4f* __restrict__ g = (const v4f*)(P + (size_t)base * CH);
      v4f* s4 = (v4f*)sp;
      const int nf4 = min(M - base, BLOCK) * CH / 4;   // normally 1920
      #pragma unroll
      for (int k = 0; k < (BLOCK * CH / 4 + BLOCK - 1) / BLOCK; ++k) {
        int j = tid + k * BLOCK;
        if (j < nf4) s4[j] = g[j];
      }
      // hint the next tile into cache (global_prefetch_b8)
      int nbase = base + NB * BLOCK;
      if (nbase + tid < M)
        __builtin_prefetch(P + (size_t)(nbase + tid) * CH, 0, 1);
    }
    __syncthreads();

    const int  i     = base + tid;
    const bool valid = (i < M);
    const int  ii    = valid ? i : 0;

    // targets: only ch4 and ch9; 120*i is 8B-aligned -> aligned b64 loads
    const v2f* __restrict__ tg = (const v2f*)(T + (size_t)ii * CH);
    v2f t45 = tg[2];
    v2f t89 = tg[4];
    float t4 = t45[0], t9 = t89[1];

    const float* p = sp + tid * CH;  // conflict-free: 30*t mod 64 is a permutation
    float x0 = p[0], y0 = p[1], w0 = p[2], h0 = p[3], c0 = p[4];
    float x1 = p[5], y1 = p[6], w1 = p[7], h1 = p[8], c1 = p[9];

    // box corners (xy/14 is a true division, matching XLA's IEEE division)
    float ax0 = x0 / 14.0f, ay0 = y0 / 14.0f;
    float ax1 = x1 / 14.0f, ay1 = y1 / 14.0f;
    float lt0x = ax0 - 0.5f * w0, lt0y = ay0 - 0.5f * h0;
    float rb0x = ax0 + 0.5f * w0, rb0y = ay0 + 0.5f * h0;
    float lt1x = ax1 - 0.5f * w1, lt1y = ay1 - 0.5f * h1;
    float rb1x = ax1 + 0.5f * w1, rb1y = ay1 + 0.5f * h1;
    // tbox == box 0, so lt2/rb2 are bit-identical to lt0/rb0.

    // b = 0 vs tbox(=box0)
    float iw0 = fmaxf(rb0x - lt0x, 0.0f);
    float ih0 = fmaxf(rb0y - lt0y, 0.0f);
    float inter0 = iw0 * ih0;
    float area10 = (rb0x - lt0x) * (rb0y - lt0y);
    float area2  = area10;  // identical computation in the reference
    float iou0   = inter0 / ((area10 + area2) - inter0);  // 1.0, or NaN if 0/0

    // b = 1 vs tbox
    float iltx = fmaxf(lt1x, lt0x), ilty = fmaxf(lt1y, lt0y);
    float irbx = fminf(rb1x, rb0x), irby = fminf(rb1y, rb0y);
    float iw1 = fmaxf(irbx - iltx, 0.0f);
    float ih1 = fmaxf(irby - ilty, 0.0f);
    float inter1 = iw1 * ih1;
    float area11 = (rb1x - lt1x) * (rb1y - lt1y);
    float iou1   = inter1 / ((area11 + area2) - inter1);

    float max_iou = jmax_nan(iou0, iou1);
    bool  sel1    = iou1 > iou0;           // argmax (first index wins ties)
    float resp_c  = sel1 ? c1 : c0;
    float nresp_c = sel1 ? c0 : c1;

    float coo = (t4 > 0.0f)  ? 1.0f : 0.0f;
    float noo = (t4 == 0.0f) ? 1.0f : 0.0f;
    float d4 = c0 - t4, d9 = c1 - t9;
    float dc = resp_c - max_iou;

    // loc_loss and class_loss are exactly zero (x - x of finite values).
    float cell = coo * (dc * dc)
               + 0.5f * (coo * (nresp_c * nresp_c) + noo * (d4 * d4 + d9 * d9));
    if (valid) acc += cell;
  }

  // wave-level exact fp32 reduction on the matrix pipe, then 8 wave sums
  float w = wave_sum_wmma(acc);
  if ((tid & 31) == 0) wred[tid >> 5] = w;
  __syncthreads();
  if (tid == 0) {
    float r = 0.0f;
    #pragma unroll
    for (int k = 0; k < WAVES; ++k) r += wred[k];
    partial[blockIdx.x] = r;
  }
}

__global__ void __launch_bounds__(BLOCK) yolo_final_kernel(
    const float* __restrict__ partial, float* __restrict__ out,
    int nb, float inv_n)
{
  __shared__ float wred[WAVES];
  float acc = 0.0f;
  for (int j = threadIdx.x; j < nb; j += BLOCK) acc += partial[j];
  float w = wave_sum_wmma(acc);
  if ((threadIdx.x & 31) == 0) wred[threadIdx.x >> 5] = w;
  __syncthreads();
  if (threadIdx.x == 0) {
    float r = 0.0f;
    #pragma unroll
    for (int k = 0; k < WAVES; ++k) r += wred[k];
    out[0] = r * inv_n;   // N = 16384 (power of two: * (1/N) == / N exactly)
  }
}

extern "C" void kernel_launch(void* const* d_in, const int* in_sizes, int n_in,
                              void* d_out, int out_size, void* d_ws, size_t ws_size,
                              hipStream_t stream)
{
  const float* P = (const float*)d_in[0];
  const float* T = (const float*)d_in[1];
  float* out     = (float*)d_out;
  float* partial = (float*)d_ws;          // NB floats = 2 KB scratch

  const int M = in_sizes[0] / CH;         // 16384 * 7 * 7 cells
  const int nBatch = M / 49;              // 16384
  const float inv_n = 1.0f / (float)nBatch;

  yolo_partial_kernel<<<NB, BLOCK, 0, stream>>>(P, T, partial, M);
  yolo_final_kernel<<<1, BLOCK, 0, stream>>>(partial, out, NB, inv_n);
}